// FuzzyTopKGrouping_60301340836401
// MI455X (gfx1250) — compile-verified
//
#include <hip/hip_runtime.h>

typedef __attribute__((ext_vector_type(2))) float v2f;
typedef __attribute__((ext_vector_type(8))) float v8f;

#define BATCH 4
#define AGENTS 8192
#define KNN 16
#define ROWS_PER_BLOCK 128          // 8 waves * 16 rows
#define WAVES_PER_BLOCK 8
#define COLS_PER_ITER 64            // 4 WMMA column tiles per sync
#define N_ITERS (AGENTS / COLS_PER_ITER)
#define BIGD2 3.0e38f

__global__ __launch_bounds__(256) void fuzzy_topk_kernel(
    const float* __restrict__ pos,
    int* __restrict__ out_idx,
    float* __restrict__ out_w)
{
    // 16 rows x 64 cols of squared distances per wave, padded to 65 floats/row
    __shared__ float lds[WAVES_PER_BLOCK][16][COLS_PER_ITER + 1];

    const int tid     = threadIdx.x;
    const int wave    = tid >> 5;
    const int lane    = tid & 31;
    const int laneLow = lane & 15;
    const int half    = lane >> 4;      // 0: lanes 0-15, 1: lanes 16-31

    const int blocksPerBatch = AGENTS / ROWS_PER_BLOCK;            // 64
    const int b              = blockIdx.x / blocksPerBatch;
    const int rowBlockBase   = (blockIdx.x % blocksPerBatch) * ROWS_PER_BLOCK;
    const int rowTileBase    = rowBlockBase + wave * 16;

    const float* posB = pos + (size_t)b * AGENTS * 3;

    // ---- A fragment (loop invariant).  32-bit 16x4 A layout:
    //  lanes 0-15: M=lane, VGPR0=K0(x), VGPR1=K1(y)
    //  lanes 16-31: M=lane-16, VGPR0=K2(z), VGPR1=K3(pad 0)
    const int ra = rowTileBase + laneLow;
    const float ax = posB[ra * 3 + 0];
    const float ay = posB[ra * 3 + 1];
    const float az = posB[ra * 3 + 2];
    v2f afrag;
    afrag.x = half ? az : ax;
    afrag.y = half ? 0.0f : ay;

    // ||row||^2 for the 8 C-matrix rows this lane holds (M = half*8 + v)
    float sqrow[8];
#pragma unroll
    for (int v = 0; v < 8; ++v) {
        const int rm = rowTileBase + half * 8 + v;
        const float x = posB[rm * 3 + 0];
        const float y = posB[rm * 3 + 1];
        const float z = posB[rm * 3 + 2];
        sqrow[v] = x * x + y * y + z * z;
    }

    // Per-row top-K state (lanes 0-15 own row rowTileBase+lane).
    // Kept as SQUARED distances, sorted ascending; sqrt applied only at the end.
    float dv[KNN];
    int   iv[KNN];
#pragma unroll
    for (int i = 0; i < KNN; ++i) { dv[i] = BIGD2; iv[i] = -1; }

    for (int it = 0; it < N_ITERS; ++it) {
        const int colIterBase = it * COLS_PER_ITER;

        // ---- 4 x WMMA over 64 candidate columns ----
#pragma unroll
        for (int t = 0; t < 4; ++t) {
            const int colBase = colIterBase + t * 16;
            const int cn = colBase + laneLow;
            const float bx = posB[cn * 3 + 0];
            const float by = posB[cn * 3 + 1];
            const float bz = posB[cn * 3 + 2];
            const float sqn = bx * bx + by * by + bz * bz;

            // B (4x16) layout mirrors A: lanes 0-15 hold (K0,K1)=(x,y) of col N,
            // lanes 16-31 hold (K2,K3)=(z,0) of col N.
            v2f bfrag;
            bfrag.x = half ? bz : bx;
            bfrag.y = half ? 0.0f : by;

            v8f c = {0.f, 0.f, 0.f, 0.f, 0.f, 0.f, 0.f, 0.f};
            c = __builtin_amdgcn_wmma_f32_16x16x4_f32(
                    /*neg_a=*/false, afrag, /*neg_b=*/false, bfrag,
                    /*c_mod=*/(short)0, c, /*reuse_a=*/false, /*reuse_b=*/false);

            // d2 = ||r||^2 + ||c||^2 - 2 r.c ; clamp at 0; mask self with BIG
#pragma unroll
            for (int v = 0; v < 8; ++v) {
                const int rm = rowTileBase + half * 8 + v;
                float d2 = sqrow[v] + sqn - 2.0f * c[v];
                d2 = fmaxf(d2, 0.0f);
                if (cn == rm) d2 = BIGD2;
                lds[wave][half * 8 + v][t * 16 + laneLow] = d2;
            }
        }

        __syncthreads();

        // ---- per-row top-K merge: lane r scans its row's 64 candidates ----
        if (lane < 16) {
#pragma unroll 8
            for (int j = 0; j < COLS_PER_ITER; ++j) {
                const float d = lds[wave][lane][j];
                if (d < dv[KNN - 1]) {          // fast reject (common case)
                    float dcur = d;
                    int   icur = colIterBase + j;
#pragma unroll
                    for (int t = 0; t < KNN; ++t) {
                        const bool sw = dcur < dv[t];
                        const float td = sw ? dv[t] : dcur;
                        const int   ti = sw ? iv[t] : icur;
                        dv[t] = sw ? dcur : dv[t];
                        iv[t] = sw ? icur : iv[t];
                        dcur = td;
                        icur = ti;
                    }
                }
            }
        }

        __syncthreads();
    }

    // ---- finalize: sqrt the 16 winners, stable softmax over -dist, store ----
    if (lane < 16) {
        const int row = rowTileBase + lane;

        float dist[KNN];
#pragma unroll
        for (int i = 0; i < KNN; ++i) dist[i] = sqrtf(dv[i]);

        const float m = dist[0];                 // min distance -> max score
        float w[KNN];
        float sum = 0.0f;
#pragma unroll
        for (int i = 0; i < KNN; ++i) {
            w[i] = expf(m - dist[i]);            // TEMPERATURE == 1.0
            sum += w[i];
        }
        const float inv = 1.0f / sum;

        const size_t base = ((size_t)b * AGENTS + row) * KNN;
#pragma unroll
        for (int i = 0; i < KNN; ++i) {
            out_idx[base + i] = iv[i];
            out_w[base + i]   = w[i] * inv;
        }
    }
}

extern "C" void kernel_launch(void* const* d_in, const int* in_sizes, int n_in,
                              void* d_out, int out_size, void* d_ws, size_t ws_size,
                              hipStream_t stream) {
    (void)in_sizes; (void)n_in; (void)out_size; (void)d_ws; (void)ws_size;

    const float* pos = (const float*)d_in[0];

    // d_out = concat(neighbor_idx[int32, B*A*K], membership[f32, B*A*K])
    int*   out_idx = (int*)d_out;
    float* out_w   = (float*)d_out + (size_t)BATCH * AGENTS * KNN;

    dim3 grid(BATCH * (AGENTS / ROWS_PER_BLOCK));   // 4 * 64 = 256 blocks
    dim3 block(256);                                // 8 wave32 per block
    hipLaunchKernelGGL(fuzzy_topk_kernel, grid, block, 0, stream,
                       pos, out_idx, out_w);
}